// GCNModelAE_55456617726628
// MI455X (gfx1250) — compile-verified
//
#include <hip/hip_runtime.h>

#define NN 16384      // N_NODES
#define NE 524288     // N_EDGES

typedef __attribute__((ext_vector_type(2))) float v2f;
typedef __attribute__((ext_vector_type(4))) float v4f;
typedef __attribute__((ext_vector_type(8))) float v8f;

__device__ __forceinline__ v8f wmma4(v2f a, v2f b, v8f c) {
    // D = A(16x4,f32) * B(4x16,f32) + C(16x16,f32)
    return __builtin_amdgcn_wmma_f32_16x16x4_f32(
        /*neg_a=*/false, a, /*neg_b=*/false, b,
        /*c_mod=*/(short)0, c, /*reuse_a=*/false, /*reuse_b=*/false);
}

// ---------------------------------------------------------------- zero fill
__global__ __launch_bounds__(256) void zero_kernel(float* __restrict__ p, long n4) {
    long i = (long)blockIdx.x * blockDim.x + threadIdx.x;
    if (i < n4) {
        v4f zv = {0.f, 0.f, 0.f, 0.f};
        ((v4f*)p)[i] = zv;
    }
}

// ---------------------------------------------------- generic f32 WMMA GEMM
// C[M,N] = (RELU_A ? relu(A) : A)[M,K] @ B[K,N], row-major.
// One wave -> 16 rows x 64 cols (4 WMMA tiles, A reused across them).
// N must be a multiple of 64, K a multiple of 4, M a multiple of 16.
template <bool RELU_A>
__global__ __launch_bounds__(256) void gemm16xN(const float* __restrict__ A,
                                                const float* __restrict__ B,
                                                float* __restrict__ C,
                                                int M, int N, int K) {
    const int wid  = threadIdx.x >> 5;
    const int lane = threadIdx.x & 31;
    const int r    = lane & 15;
    const int hi   = lane >> 4;               // 0 or 1

    const int jGroups = N >> 6;               // groups of 64 columns
    const int gw      = blockIdx.x * 8 + wid; // global wave id
    const int iTile   = gw / jGroups;
    const int jG      = gw - iTile * jGroups;
    if (iTile * 16 >= M) return;              // wave-uniform

    const float* Arow = A + (size_t)(iTile * 16 + r) * K + 2 * hi;
    const float* Bcol = B + (size_t)jG * 64 + r;

    v8f c0 = {}, c1 = {}, c2 = {}, c3 = {};
    for (int kc = 0; kc < K; kc += 4) {
        v2f a = *(const v2f*)(Arow + kc);
        if (RELU_A) { a.x = fmaxf(a.x, 0.f); a.y = fmaxf(a.y, 0.f); }
        const float* B0 = Bcol + (size_t)(kc + 2 * hi) * N;
        const float* B1 = B0 + N;
        v2f b;
        b.x = B0[0];  b.y = B1[0];  c0 = wmma4(a, b, c0);
        b.x = B0[16]; b.y = B1[16]; c1 = wmma4(a, b, c1);
        b.x = B0[32]; b.y = B1[32]; c2 = wmma4(a, b, c2);
        b.x = B0[48]; b.y = B1[48]; c3 = wmma4(a, b, c3);
    }

    float* Crow = C + (size_t)(iTile * 16 + 8 * hi) * N + (size_t)jG * 64 + r;
#pragma unroll
    for (int v = 0; v < 8; ++v) {
        Crow[(size_t)v * N + 0]  = c0[v];
        Crow[(size_t)v * N + 16] = c1[v];
        Crow[(size_t)v * N + 32] = c2[v];
        Crow[(size_t)v * N + 48] = c3[v];
    }
}

// --------------------------------------------------------- adj = z @ z^T
// z: [NN, 64]. The transposed B read collapses to the same per-lane
// contiguous-float2 pattern as A. grid = (32, 1024), block = 256.
__global__ __launch_bounds__(256) void gemm_zzt(const float* __restrict__ z,
                                                float* __restrict__ adj) {
    const int wid  = threadIdx.x >> 5;
    const int lane = threadIdx.x & 31;
    const int r    = lane & 15;
    const int hi   = lane >> 4;
    const int iTile = blockIdx.y;                 // 0..1023 (row tile)
    const int jG    = blockIdx.x * 8 + wid;       // 0..255 (64-col group)

    const float* Arow = z + (size_t)(iTile * 16 + r) * 64 + 2 * hi;
    v2f a[16];
#pragma unroll
    for (int kc = 0; kc < 16; ++kc) a[kc] = *(const v2f*)(Arow + kc * 4);

#pragma unroll
    for (int t = 0; t < 4; ++t) {
        const float* Brow = z + (size_t)(jG * 64 + t * 16 + r) * 64 + 2 * hi;
        v8f c = {};
#pragma unroll
        for (int kc = 0; kc < 16; ++kc) {
            v2f b = *(const v2f*)(Brow + kc * 4);
            c = wmma4(a[kc], b, c);
        }
        float* Crow = adj + (size_t)(iTile * 16 + 8 * hi) * NN
                          + (size_t)jG * 64 + t * 16 + r;
#pragma unroll
        for (int v = 0; v < 8; ++v) Crow[(size_t)v * NN] = c[v];
    }
}

// ------------------------------------------------------------- SpMM scatter
// out[dst] += w * H[src]; one block covers EPB edges, D features each.
template <int D, int EPB>
__global__ __launch_bounds__(256) void spmm_scatter(const float* __restrict__ H,
                                                    const int* __restrict__ es,
                                                    const int* __restrict__ ed,
                                                    const float* __restrict__ ew,
                                                    float* __restrict__ out) {
    const int e = blockIdx.x * EPB + threadIdx.x / D;
    const int f = threadIdx.x & (D - 1);
    const int s = es[e];
    const int d = ed[e];
    const float w = ew[e];
    atomicAdd(out + (size_t)d * D + f, w * H[(size_t)s * D + f]);
}

// ------------------------------------------------------------------- launch
extern "C" void kernel_launch(void* const* d_in, const int* in_sizes, int n_in,
                              void* d_out, int out_size, void* d_ws, size_t ws_size,
                              hipStream_t stream) {
    const float* x  = (const float*)d_in[0];
    const int*   es = (const int*)d_in[1];
    const int*   ed = (const int*)d_in[2];
    const float* ew = (const float*)d_in[3];
    const float* W1 = (const float*)d_in[4];
    const float* W2 = (const float*)d_in[5];

    float* adj  = (float*)d_out;                        // [NN, NN]
    float* zout = (float*)d_out + (size_t)NN * NN;      // [NN, 64]

    float* xw1 = (float*)d_ws;                          // [NN, 256]
    float* h1  = xw1 + (size_t)NN * 256;                // [NN, 256]
    float* t2  = h1  + (size_t)NN * 256;                // [NN, 64]

    // zero the scatter destinations (ws/out are poisoned, not re-poisoned)
    {
        long n4 = (long)NN * 256 / 4;
        zero_kernel<<<(int)((n4 + 255) / 256), 256, 0, stream>>>(h1, n4);
        long z4 = (long)NN * 64 / 4;
        zero_kernel<<<(int)((z4 + 255) / 256), 256, 0, stream>>>(zout, z4);
    }

    // 1) xw1 = x @ W1              (M=16384, K=512, N=256) -> 4096 waves
    gemm16xN<false><<<512, 256, 0, stream>>>(x, W1, xw1, NN, 256, 512);

    // 2) h1 = spmm(edges, xw1)     (relu deferred to next GEMM's A-load)
    spmm_scatter<256, 1><<<NE, 256, 0, stream>>>(xw1, es, ed, ew, h1);

    // 3) t2 = relu(h1) @ W2        (M=16384, K=256, N=64) -> 1024 waves
    gemm16xN<true><<<128, 256, 0, stream>>>(h1, W2, t2, NN, 64, 256);

    // 4) z = spmm(edges, t2)
    spmm_scatter<64, 4><<<NE / 4, 256, 0, stream>>>(t2, es, ed, ew, zout);

    // 5) adj = z @ z^T             (bandwidth-bound: 1 GiB store)
    dim3 g(32, 1024);
    gemm_zzt<<<g, 256, 0, stream>>>(zout, adj);
}